// LightGCNMModel_28157805592960
// MI455X (gfx1250) — compile-verified
//
#include <hip/hip_runtime.h>
#include <hip/hip_bf16.h>

#define NU      100000
#define NI      50000
#define NNODES  150000
#define EK      64
#define FEAT    1024
#define NE      3000000
#define BATCH   16384
#define NLAYERS 3

typedef float v2f __attribute__((ext_vector_type(2)));
typedef float v8f __attribute__((ext_vector_type(8)));

// vector-of-4-int pointer types matching the async-to-LDS builtin signature
typedef int v4i_vs __attribute__((vector_size(4 * sizeof(int))));
typedef __attribute__((address_space(1))) v4i_vs* gptr_v4i;  // global
typedef __attribute__((address_space(3))) v4i_vs* lptr_v4i;  // LDS

// Native CDNA5 fp32 atomic add, device scope, no return (STOREcnt-tracked;
// drained by the implicit wait-idle at s_endpgm).
__device__ __forceinline__ void atomic_add_f32_dev(float* p, float v) {
    asm volatile("global_atomic_add_f32 %0, %1, off scope:SCOPE_DEV"
                 :: "v"((unsigned long long)(size_t)p), "v"(v)
                 : "memory");
}

// ---------------------------------------------------------------- utilities
__global__ void lgcn_zero4_kernel(float4* __restrict__ p, int n4) {
    int i = blockIdx.x * blockDim.x + threadIdx.x;
    if (i < n4) p[i] = make_float4(0.f, 0.f, 0.f, 0.f);
}

// deg[dst] += 1 for both edge directions (deg stored in `dinv` buffer)
__global__ void lgcn_degree_kernel(const int* __restrict__ ue,
                                   const int* __restrict__ ie,
                                   float* __restrict__ deg) {
    int e = blockIdx.x * blockDim.x + threadIdx.x;
    if (e < NE) {
        atomic_add_f32_dev(&deg[ue[e]], 1.0f);
        atomic_add_f32_dev(&deg[NU + ie[e]], 1.0f);
    }
}

// deg -> dinv in place: deg>0 ? rsqrt(max(deg,1)) : 0
__global__ void lgcn_dinv_kernel(float* __restrict__ deg) {
    int i = blockIdx.x * blockDim.x + threadIdx.x;
    if (i < NNODES) {
        float d = deg[i];
        deg[i] = (d > 0.0f) ? rsqrtf(fmaxf(d, 1.0f)) : 0.0f;
    }
}

// per-original-edge norm (same value serves both directions)
__global__ void lgcn_norm_kernel(const int* __restrict__ ue,
                                 const int* __restrict__ ie,
                                 const float* __restrict__ dinv,
                                 float* __restrict__ nrm) {
    int e = blockIdx.x * blockDim.x + threadIdx.x;
    if (e < NE) nrm[e] = dinv[ue[e]] * dinv[NU + ie[e]];
}

// x0 = concat(Gu, Gi); acc = x0
__global__ void lgcn_init_kernel(const float* __restrict__ Gu,
                                 const float* __restrict__ Gi,
                                 float* __restrict__ x,
                                 float* __restrict__ acc) {
    int i = blockIdx.x * blockDim.x + threadIdx.x;
    if (i < NNODES * EK) {
        float v = (i < NU * EK) ? Gu[i] : Gi[i - NU * EK];
        x[i] = v;
        acc[i] = v;
    }
}

// one LGConv layer: y[item] += x[user]*w ; y[user] += x[item]*w
// thread = (edge, float2 component pair); 32 threads per edge, both directions
__global__ void lgcn_scatter_kernel(const float* __restrict__ x,
                                    float* __restrict__ y,
                                    const int* __restrict__ ue,
                                    const int* __restrict__ ie,
                                    const float* __restrict__ nrm) {
    int gid = blockIdx.x * blockDim.x + threadIdx.x;  // < NE*32 = 96,000,000
    int e  = gid >> 5;
    int c2 = (gid & 31) * 2;
    int u  = ue[e];
    int it = ie[e];
    float w = nrm[e];
    v2f xu = *(const v2f*)(x + u * EK + c2);
    v2f xi = *(const v2f*)(x + (NU + it) * EK + c2);
    float* yi = y + (NU + it) * EK + c2;
    float* yu = y + u * EK + c2;
    atomic_add_f32_dev(yi,     xu.x * w);
    atomic_add_f32_dev(yi + 1, xu.y * w);
    atomic_add_f32_dev(yu,     xi.x * w);
    atomic_add_f32_dev(yu + 1, xi.y * w);
}

__global__ void lgcn_accum4_kernel(float4* __restrict__ acc,
                                   const float4* __restrict__ y, int n4) {
    int i = blockIdx.x * blockDim.x + threadIdx.x;
    if (i < n4) {
        float4 a = acc[i];
        float4 b = y[i];
        acc[i] = make_float4(a.x + b.x, a.y + b.y, a.z + b.z, a.w + b.w);
    }
}

// ---------------------------------------------------------------- WMMA GEMM
// proj[b][n] = sum_k F[items[b]][k] * W[n][k] + bias[n]
// 1 WG = 4 waves = 16 batch rows x 64 output cols; fp32 WMMA 16x16x4.
__global__ __launch_bounds__(128)
void lgcn_proj_gemm_kernel(const float* __restrict__ F,
                           const float* __restrict__ W,
                           const float* __restrict__ bias,
                           const int* __restrict__ items,
                           float* __restrict__ proj) {
    // 16 rows x 64 K-chunk of gathered F; +4 float row pad to spread LDS banks
    __shared__ float ldsF[16][68];

    const int tid  = threadIdx.x;
    const int wave = tid >> 5;         // 0..3 -> owns output cols [wave*16, +16)
    const int lane = tid & 31;
    const int bm   = blockIdx.x * 16;  // batch row base
    const int row  = lane & 15;        // A: M index / B: N index within tile
    const int kp   = (lane >> 4) << 1; // lanes 0-15 -> K=0,1 ; lanes 16-31 -> K=2,3
    const int ncol = wave * 16 + row;

    const float* Wrow = W + (size_t)ncol * FEAT;

    v8f c = {};

    for (int k0 = 0; k0 < FEAT; k0 += 64) {
        // cooperative gather of F tile: 128 threads x 16 B x 2, async to LDS
        {
            int r  = tid >> 3;        // 0..15
            int c4 = (tid & 7) * 8;   // 0..56
            const float* src = F + (size_t)items[bm + r] * FEAT + k0 + c4;
            float* dst = &ldsF[r][c4];
#if __has_builtin(__builtin_amdgcn_global_load_async_to_lds_b128)
            __builtin_amdgcn_global_load_async_to_lds_b128(
                (gptr_v4i)src, (lptr_v4i)dst, 0, 0);
            __builtin_amdgcn_global_load_async_to_lds_b128(
                (gptr_v4i)(src + 4), (lptr_v4i)(dst + 4), 0, 0);
  #if __has_builtin(__builtin_amdgcn_s_wait_asynccnt)
            __builtin_amdgcn_s_wait_asynccnt(0);
  #else
            asm volatile("s_wait_asynccnt 0x0" ::: "memory");
  #endif
#else
            float4 v0 = *(const float4*)(src);
            float4 v1 = *(const float4*)(src + 4);
            *(float4*)dst       = v0;
            *(float4*)(dst + 4) = v1;
#endif
        }
        __syncthreads();

        #pragma unroll
        for (int kk = 0; kk < 64; kk += 4) {
            v2f a = *(const v2f*)(&ldsF[row][kk + kp]);     // A[M=row][K=kp,kp+1]
            v2f b = *(const v2f*)(Wrow + k0 + kk + kp);     // B[K=kp,kp+1][N=row]
            // D = A(16x4) x B(4x16) + C, fp32
            c = __builtin_amdgcn_wmma_f32_16x16x4_f32(
                    false, a, false, b, (short)0, c, false, false);
        }
        __syncthreads();
    }

    // D layout: lanes 0-15 -> N=lane, VGPR g -> M=g ; lanes 16-31 -> M=g+8
    const float bn   = bias[ncol];
    const int mbase  = (lane >> 4) * 8;
    #pragma unroll
    for (int g = 0; g < 8; ++g) {
        int m = mbase + g;
        proj[(size_t)(bm + m) * EK + ncol] = c[g] + bn;
    }
}

// ---------------------------------------------------------------- final dots
// one wave per batch element: out[b] = (acc_u . acc_i)/16 + (Tu_u . proj_b/||proj_b||)
__global__ void lgcn_final_kernel(const float* __restrict__ acc,
                                  const float* __restrict__ Tu,
                                  const float* __restrict__ proj,
                                  const int* __restrict__ users,
                                  const int* __restrict__ items,
                                  float* __restrict__ out) {
    int b    = blockIdx.x * 8 + (threadIdx.x >> 5);
    int lane = threadIdx.x & 31;
    int u  = users[b];
    int it = items[b];
    const float* au = acc  + (size_t)u * EK;
    const float* ai = acc  + (size_t)(NU + it) * EK;
    const float* tu = Tu   + (size_t)u * EK;
    const float* pr = proj + (size_t)b * EK;

    float d1 = au[lane] * ai[lane] + au[lane + 32] * ai[lane + 32];
    float p0 = pr[lane], p1 = pr[lane + 32];
    float nn = p0 * p0 + p1 * p1;
    float d2 = tu[lane] * p0 + tu[lane + 32] * p1;

    #pragma unroll
    for (int off = 16; off; off >>= 1) {
        d1 += __shfl_xor(d1, off, 32);
        nn += __shfl_xor(nn, off, 32);
        d2 += __shfl_xor(d2, off, 32);
    }
    if (lane == 0) {
        float inv = 1.0f / fmaxf(sqrtf(nn), 1e-12f);
        out[b] = d1 * (1.0f / 16.0f) + d2 * inv;
    }
}

// ---------------------------------------------------------------- launcher
extern "C" void kernel_launch(void* const* d_in, const int* in_sizes, int n_in,
                              void* d_out, int out_size, void* d_ws, size_t ws_size,
                              hipStream_t stream) {
    (void)in_sizes; (void)n_in; (void)out_size; (void)ws_size;

    const float* Gu = (const float*)d_in[0];
    const float* Gi = (const float*)d_in[1];
    const float* Tu = (const float*)d_in[2];
    const float* F  = (const float*)d_in[3];
    const float* Pw = (const float*)d_in[4];
    const float* Pb = (const float*)d_in[5];
    const int* ue    = (const int*)d_in[6];
    const int* ie    = (const int*)d_in[7];
    const int* users = (const int*)d_in[8];
    const int* items = (const int*)d_in[9];
    float* out = (float*)d_out;

    const int NK = NNODES * EK;            // 9,600,000 floats
    float* ws   = (float*)d_ws;
    float* acc  = ws;                      // NK
    float* bufA = acc  + NK;               // NK
    float* bufB = bufA + NK;               // NK
    float* dinv = bufB + NK;               // NNODES
    float* nrm  = dinv + NNODES;           // NE
    float* proj = nrm  + NE;               // BATCH*EK

    const dim3 T(256);
    auto nb = [](int n) { return dim3((unsigned)((n + 255) / 256)); };

    // degrees -> dinv -> per-edge norms
    lgcn_zero4_kernel <<<nb(NNODES / 4), T, 0, stream>>>((float4*)dinv, NNODES / 4);
    lgcn_degree_kernel<<<nb(NE),         T, 0, stream>>>(ue, ie, dinv);
    lgcn_dinv_kernel  <<<nb(NNODES),     T, 0, stream>>>(dinv);
    lgcn_norm_kernel  <<<nb(NE),         T, 0, stream>>>(ue, ie, dinv, nrm);

    // x0 = concat(Gu,Gi); acc = x0
    lgcn_init_kernel<<<nb(NK), T, 0, stream>>>(Gu, Gi, bufA, acc);

    // 3 LGConv layers, ping-pong
    float* x = bufA;
    float* y = bufB;
    for (int l = 0; l < NLAYERS; ++l) {
        lgcn_zero4_kernel  <<<nb(NK / 4), T, 0, stream>>>((float4*)y, NK / 4);
        lgcn_scatter_kernel<<<dim3(NE * 32 / 256), T, 0, stream>>>(x, y, ue, ie, nrm);
        lgcn_accum4_kernel <<<nb(NK / 4), T, 0, stream>>>((float4*)acc, (const float4*)y, NK / 4);
        float* t = x; x = y; y = t;
    }

    // gather-GEMM through WMMA, then fused norms + dots
    lgcn_proj_gemm_kernel<<<dim3(BATCH / 16), dim3(128), 0, stream>>>(F, Pw, Pb, items, proj);
    lgcn_final_kernel<<<dim3(BATCH / 8), T, 0, stream>>>(acc, Tu, proj, users, items, out);
}